// MeshRendererWithDepth2_35862976922349
// MI455X (gfx1250) — compile-verified
//
#include <hip/hip_runtime.h>
#include <stdint.h>

// ---------------- problem constants ----------------
#define V_CNT   6890
#define FC_CNT  13776
#define K_FACE  28
#define NPIX    (2*336*336)          // 225792 pixels

#define SIGMA_INV  10000.0f          // 1/1e-4
#define GAMMA_INV  10000.0f          // 1/1e-4
#define EPS_BLEND  1e-10f
#define ZFAR       100.0f
#define ZRANGE_INV (1.0f/(100.0f-1.0f))

// packed per-vertex record: pos(3) | normal(3) | color(3) = 9 floats
#define PACKED_FLOATS     (V_CNT*9)                 // 62010
#define PACKED_CHUNKS     ((PACKED_FLOATS+3)/4)     // float4 chunks = 15503
#define PACKED_PAD_FLOATS (PACKED_CHUNKS*4)         // 62012
#define PACKED_BYTES      (PACKED_CHUNKS*16)        // 248048 B  (< 320KB WGP LDS)

typedef int v4i_t __attribute__((ext_vector_type(4)));

__device__ __forceinline__ float fast_rcp(float x) {
#if defined(__AMDGCN__) && __has_builtin(__builtin_amdgcn_rcpf)
  return __builtin_amdgcn_rcpf(x);
#else
  return 1.f / x;
#endif
}
__device__ __forceinline__ float fast_rsqrt(float x) {
#if defined(__AMDGCN__) && __has_builtin(__builtin_amdgcn_rsqf)
  return __builtin_amdgcn_rsqf(x);
#else
  return rsqrtf(x);
#endif
}

// ---------------- geometry pre-pass ----------------
__global__ void init_packed_kernel(const float* __restrict__ verts,
                                   const float* __restrict__ vcols,
                                   float* __restrict__ packed) {
  int v = blockIdx.x * blockDim.x + threadIdx.x;
  if (v < V_CNT) {
    float* p = packed + v * 9;
    p[0] = verts[v*3+0]; p[1] = verts[v*3+1]; p[2] = verts[v*3+2];
    p[3] = 0.f;          p[4] = 0.f;          p[5] = 0.f;      // normal accumulator
    p[6] = vcols[v*3+0]; p[7] = vcols[v*3+1]; p[8] = vcols[v*3+2];
  }
  int tail = v - V_CNT;
  if (tail >= 0 && tail < (PACKED_PAD_FLOATS - PACKED_FLOATS))
    packed[PACKED_FLOATS + tail] = 0.f;
}

__global__ void accum_face_normals_kernel(const float* __restrict__ verts,
                                          const int* __restrict__ faces,
                                          float* __restrict__ packed) {
  int f = blockIdx.x * blockDim.x + threadIdx.x;
  if (f >= FC_CNT) return;
  int i0 = faces[f*3+0], i1 = faces[f*3+1], i2 = faces[f*3+2];
  float ax = verts[i0*3+0], ay = verts[i0*3+1], az = verts[i0*3+2];
  float bx = verts[i1*3+0], by = verts[i1*3+1], bz = verts[i1*3+2];
  float cx = verts[i2*3+0], cy = verts[i2*3+1], cz = verts[i2*3+2];
  float e1x = bx-ax, e1y = by-ay, e1z = bz-az;
  float e2x = cx-ax, e2y = cy-ay, e2z = cz-az;
  float nx = e1y*e2z - e1z*e2y;
  float ny = e1z*e2x - e1x*e2z;
  float nz = e1x*e2y - e1y*e2x;
  int idx[3] = {i0, i1, i2};
#pragma unroll
  for (int j = 0; j < 3; ++j) {
    float* p = packed + idx[j]*9 + 3;
    atomicAdd(p+0, nx); atomicAdd(p+1, ny); atomicAdd(p+2, nz);
  }
}

__global__ void normalize_vnorm_kernel(float* __restrict__ packed) {
  int v = blockIdx.x * blockDim.x + threadIdx.x;
  if (v >= V_CNT) return;
  float* p = packed + v*9 + 3;
  float nx = p[0], ny = p[1], nz = p[2];
  // 1/max(sqrt(x),1e-6) == rsqrt(max(x,1e-12))
  float s = fast_rsqrt(fmaxf(nx*nx + ny*ny + nz*nz, 1e-12f));
  p[0] = nx*s; p[1] = ny*s; p[2] = nz*s;
}

// ---------------- main per-pixel renderer ----------------
__global__ __launch_bounds__(1024)
void render_kernel(const int*   __restrict__ faces,
                   const int*   __restrict__ p2f,
                   const float* __restrict__ bary,
                   const float* __restrict__ dists,
                   const float* __restrict__ zbuf,
                   const float* __restrict__ packed,
                   const float* __restrict__ light,
                   const float* __restrict__ amb,
                   const float* __restrict__ dif,
                   const float* __restrict__ spc,
                   const float* __restrict__ cam,
                   float* __restrict__ out_img) {
  extern __shared__ float4 lds4[];
  float* ldsF = (float*)lds4;
  const float4* g4 = (const float4*)packed;

  // ---- stage packed vertex table into LDS via CDNA5 async global->LDS DMA ----
#if defined(__gfx1250__) && __has_builtin(__builtin_amdgcn_global_load_async_to_lds_b128)
  for (int i = threadIdx.x; i < PACKED_CHUNKS; i += blockDim.x) {
    __builtin_amdgcn_global_load_async_to_lds_b128(
        (__attribute__((address_space(1))) v4i_t*)(uintptr_t)(g4 + i),
        (__attribute__((address_space(3))) v4i_t*)(uint32_t)(uintptr_t)(lds4 + i),
        /*offset=*/0, /*cpol=*/0);
  }
  #if __has_builtin(__builtin_amdgcn_s_wait_asynccnt)
  __builtin_amdgcn_s_wait_asynccnt(0);
  #else
  asm volatile("s_wait_asynccnt 0" ::: "memory");
  #endif
#else
  for (int i = threadIdx.x; i < PACKED_CHUNKS; i += blockDim.x) lds4[i] = g4[i];
#endif
  __syncthreads();

  int pix = blockIdx.x * blockDim.x + threadIdx.x;
  if (pix >= NPIX) return;

  const int base = pix * K_FACE;               // 28 elems: 112B, 16B aligned
  const v4i_t*  p2f4 = (const v4i_t*) (p2f   + base);
  const float4* zb4p = (const float4*)(zbuf  + base);
  const float4* dd4p = (const float4*)(dists + base);
  const float4* ba4p = (const float4*)(bary  + (size_t)base * 3);  // 336B, aligned

  __builtin_prefetch(ba4p, 0, 0);
  __builtin_prefetch(dd4p, 0, 0);

  // uniforms (scalar loads)
  float lx = light[0], ly = light[1], lz = light[2];
  float ar = amb[0],  ag = amb[1],  ab = amb[2];
  float dr = dif[0],  dg = dif[1],  db = dif[2];
  float sr = spc[0],  sg = spc[1],  sb = spc[2];
  float cxp = cam[0], cyp = cam[1], czp = cam[2];

  // ---- pass 1: z_inv_max (vectorized b128 loads) ----
  float zmax = 0.f;
#pragma unroll
  for (int g = 0; g < 7; ++g) {
    v4i_t  pf = p2f4[g];
    float4 zb = zb4p[g];
    zmax = fmaxf(zmax, (pf.x >= 0) ? (ZFAR - zb.x) * ZRANGE_INV : 0.f);
    zmax = fmaxf(zmax, (pf.y >= 0) ? (ZFAR - zb.y) * ZRANGE_INV : 0.f);
    zmax = fmaxf(zmax, (pf.z >= 0) ? (ZFAR - zb.z) * ZRANGE_INV : 0.f);
    zmax = fmaxf(zmax, (pf.w >= 0) ? (ZFAR - zb.w) * ZRANGE_INV : 0.f);
  }
  zmax = fmaxf(zmax, EPS_BLEND);

  // ---- pass 2: shade + blend ----
  float sumw = 0.f, aR = 0.f, aG = 0.f, aB = 0.f, pOM = 1.f;

  auto sample = [&](int pf, float b0, float b1, float b2, float zb, float d) {
    float m = (pf >= 0) ? 1.f : 0.f;
    int idx = (pf >= 0) ? pf : 0;

    int f0 = faces[idx*3+0], f1 = faces[idx*3+1], f2 = faces[idx*3+2];
    const float* v0 = &ldsF[f0*9];
    const float* v1 = &ldsF[f1*9];
    const float* v2 = &ldsF[f2*9];

    float px = b0*v0[0] + b1*v1[0] + b2*v2[0];
    float py = b0*v0[1] + b1*v1[1] + b2*v2[1];
    float pz = b0*v0[2] + b1*v1[2] + b2*v2[2];

    float nx = b0*v0[3] + b1*v1[3] + b2*v2[3];
    float ny = b0*v0[4] + b1*v1[4] + b2*v2[4];
    float nz = b0*v0[5] + b1*v1[5] + b2*v2[5];
    float ns = fast_rsqrt(fmaxf(nx*nx + ny*ny + nz*nz, 1e-12f));
    nx *= ns; ny *= ns; nz *= ns;

    float tr = b0*v0[6] + b1*v1[6] + b2*v2[6];
    float tg = b0*v0[7] + b1*v1[7] + b2*v2[7];
    float tb = b0*v0[8] + b1*v1[8] + b2*v2[8];

    // light direction
    float ox = lx - px, oy = ly - py, oz = lz - pz;
    float os = fast_rsqrt(fmaxf(ox*ox + oy*oy + oz*oz, 1e-12f));
    ox *= os; oy *= os; oz *= os;
    float ldn = ox*nx + oy*ny + oz*nz;
    float ndotl = fmaxf(ldn, 0.f);

    // view direction
    float wx = cxp - px, wy = cyp - py, wz = czp - pz;
    float ws = fast_rsqrt(fmaxf(wx*wx + wy*wy + wz*wz, 1e-12f));
    wx *= ws; wy *= ws; wz *= ws;

    // reflect = -l + 2*(l.n)*n ; cos_a^64 via 6 squarings
    float rx = -ox + 2.f*ldn*nx;
    float ry = -oy + 2.f*ldn*ny;
    float rz = -oz + 2.f*ldn*nz;
    float ca = fmaxf(rx*wx + ry*wy + rz*wz, 0.f);
    float c2 = ca*ca, c4 = c2*c2, c8 = c4*c4, c16 = c8*c8, c32 = c16*c16, c64 = c32*c32;

    float colR = (ar + dr*ndotl)*tr + sr*c64;
    float colG = (ag + dg*ndotl)*tg + sg*c64;
    float colB = (ab + db*ndotl)*tb + sb*c64;

    float prob = m * fast_rcp(1.f + __expf(d * SIGMA_INV));  // sigmoid(-d/sigma)*mask
    float z = m * ((ZFAR - zb) * ZRANGE_INV);
    float w = prob * __expf((z - zmax) * GAMMA_INV);

    sumw += w;
    aR += w*colR; aG += w*colG; aB += w*colB;
    pOM *= (1.f - prob);
  };

  for (int g = 0; g < 7; ++g) {
    v4i_t  pf = p2f4[g];
    float4 zb = zb4p[g];
    float4 dd = dd4p[g];
    float4 ba = ba4p[3*g+0];
    float4 bb = ba4p[3*g+1];
    float4 bc = ba4p[3*g+2];
    sample(pf.x, ba.x, ba.y, ba.z, zb.x, dd.x);
    sample(pf.y, ba.w, bb.x, bb.y, zb.y, dd.y);
    sample(pf.z, bb.z, bb.w, bc.x, zb.z, dd.z);
    sample(pf.w, bc.y, bc.z, bc.w, zb.w, dd.w);
  }

  float delta = __expf((EPS_BLEND - zmax) * GAMMA_INV);
  float inv = 1.f / (sumw + delta);               // once per pixel: keep exact
  out_img[(size_t)pix*4+0] = (aR + delta) * inv;  // BG = (1,1,1)
  out_img[(size_t)pix*4+1] = (aG + delta) * inv;
  out_img[(size_t)pix*4+2] = (aB + delta) * inv;
  out_img[(size_t)pix*4+3] = 1.f - pOM;
}

// ---------------- host launcher ----------------
extern "C" void kernel_launch(void* const* d_in, const int* in_sizes, int n_in,
                              void* d_out, int out_size, void* d_ws, size_t ws_size,
                              hipStream_t stream) {
  const float* verts = (const float*)d_in[0];
  const float* vcols = (const float*)d_in[1];
  const int*   faces = (const int*)  d_in[2];
  const int*   p2f   = (const int*)  d_in[3];
  const float* bary  = (const float*)d_in[4];
  const float* dists = (const float*)d_in[5];
  const float* zbuf  = (const float*)d_in[6];
  const float* light = (const float*)d_in[7];
  const float* amb   = (const float*)d_in[8];
  const float* dif   = (const float*)d_in[9];
  const float* spc   = (const float*)d_in[10];
  const float* cam   = (const float*)d_in[11];
  float* out    = (float*)d_out;
  float* packed = (float*)d_ws;   // PACKED_BYTES of scratch

  init_packed_kernel<<<(V_CNT + 255)/256 + 1, 256, 0, stream>>>(verts, vcols, packed);
  accum_face_normals_kernel<<<(FC_CNT + 255)/256, 256, 0, stream>>>(verts, faces, packed);
  normalize_vnorm_kernel<<<(V_CNT + 255)/256, 256, 0, stream>>>(packed);

  int blocks = (NPIX + 1023) / 1024;
  render_kernel<<<blocks, 1024, PACKED_BYTES, stream>>>(
      faces, p2f, bary, dists, zbuf, packed, light, amb, dif, spc, cam, out);

  // zbuf pass-through: second output, concatenated after images
  (void)hipMemcpyAsync(out + (size_t)NPIX*4, zbuf, (size_t)NPIX*K_FACE*sizeof(float),
                       hipMemcpyDeviceToDevice, stream);
}